// MultiBoxLoss_34720515621405
// MI455X (gfx1250) — compile-verified
//
#include <hip/hip_runtime.h>

// ---------------------------------------------------------------------------
// MultiBoxLoss for MI455X (gfx1250, wave32).
// B=128 batches, P=8732 priors, C=21 classes, O=8 truths.
// Memory-bound (~112MB streamed -> ~5us roofline at 23.3TB/s).
//  - conf rows double-buffered into LDS via async global->LDS b128 DMA
//    (ASYNCcnt), overlapping next-tile fetch with current-tile compute
//  - row-sum of exp() on the matrix pipe (V_WMMA_F32_16X16X4_F32, fp32 exact)
//  - hard-negative mining via exact 4-pass radix select (no sort)
// ---------------------------------------------------------------------------

#define Bq 128
#define Pq 8732
#define Cq 21
#define Oq 8
#define TPB 256

typedef __attribute__((ext_vector_type(2))) float v2f;
typedef __attribute__((ext_vector_type(8))) float v8f;
typedef __attribute__((ext_vector_type(4))) int   v4i;

#if defined(__gfx1250__) && __has_builtin(__builtin_amdgcn_wmma_f32_16x16x4_f32)
#define USE_WMMA 1
#else
#define USE_WMMA 0
#endif

#if defined(__gfx1250__) && \
    __has_builtin(__builtin_amdgcn_global_load_async_to_lds_b128) && \
    __has_builtin(__builtin_amdgcn_s_wait_asynccnt)
#define USE_ASYNC 1
#else
#define USE_ASYNC 0
#endif

// ---------------------------------------------------------------------------
// Kernel 1: per-batch match + encode + smooth-L1 + cross entropy.
// Writes per-block partials (deterministic) and the ce_neg array for mining.
// ---------------------------------------------------------------------------
__global__ __launch_bounds__(TPB) void mbx_match_ce(
    const float* __restrict__ loc, const float* __restrict__ conf,
    const float* __restrict__ dbox, const float* __restrict__ targets,
    float* __restrict__ partL, float* __restrict__ partC,
    int* __restrict__ numpos, float* __restrict__ ceneg)
{
  const int b = blockIdx.x, tid = threadIdx.x;
  const int wave = tid >> 5, lane = tid & 31;

  __shared__ float s_truth[Oq][5];
  __shared__ unsigned long long s_bp[Oq];
  __shared__ int s_bpidx[Oq];
  __shared__ __align__(16) float s_conf[2][TPB * Cq];  // 2 x 21504B tiles
  __shared__ float s_rsum[8][32];                      // per-wave WMMA rowsums
  __shared__ float s_red[TPB];

  if (tid < Oq * 5) ((float*)s_truth)[tid] = targets[b * Oq * 5 + tid];
  if (tid < Oq) s_bp[tid] = 0ull;
  __syncthreads();

  // -------- Phase 1: per-truth best prior (argmax over P, first-tie) --------
  float bestv[Oq];
  int   bestp[Oq];
#pragma unroll
  for (int j = 0; j < Oq; ++j) { bestv[j] = -1.f; bestp[j] = 0; }

  for (int p = tid; p < Pq; p += TPB) {
    const float cx = dbox[p*4+0], cy = dbox[p*4+1];
    const float w  = dbox[p*4+2], h  = dbox[p*4+3];
    const float px0 = cx - 0.5f*w, py0 = cy - 0.5f*h;
    const float px1 = cx + 0.5f*w, py1 = cy + 0.5f*h;
    const float ab = w * h;
#pragma unroll
    for (int j = 0; j < Oq; ++j) {
      const float tx0 = s_truth[j][0], ty0 = s_truth[j][1];
      const float tx1 = s_truth[j][2], ty1 = s_truth[j][3];
      const float iw = fmaxf(fminf(tx1,px1) - fmaxf(tx0,px0), 0.f);
      const float ih = fmaxf(fminf(ty1,py1) - fmaxf(ty0,py0), 0.f);
      const float inter = iw * ih;
      const float aa = (tx1-tx0) * (ty1-ty0);
      const float iou = inter / (aa + ab - inter);
      if (iou > bestv[j]) { bestv[j] = iou; bestp[j] = p; }
    }
  }
  // pack (iou_bits<<32) | (~p) so max() == highest iou, lowest p on tie
#pragma unroll
  for (int j = 0; j < Oq; ++j) {
    unsigned long long pk =
        ((unsigned long long)__float_as_uint(bestv[j]) << 32) |
        (unsigned long long)(0xFFFFFFFFu - (unsigned)bestp[j]);
    atomicMax(&s_bp[j], pk);
  }
  __syncthreads();
  if (tid < Oq)
    s_bpidx[tid] = (int)(0xFFFFFFFFu - (unsigned)(s_bp[tid] & 0xFFFFFFFFull));
  __syncthreads();

  // -------- Phase 2: tiles of 256 priors, async double-buffered LDS --------
  float lacc = 0.f, cacc = 0.f;
  int np = 0;
  const size_t confbase = (size_t)b * Pq * Cq;
  const int ntiles = (Pq + TPB - 1) / TPB;

#if USE_ASYNC
  {
    // prologue: kick off tile 0 (all tile dword counts are multiples of 4,
    // all tile bases 16B-aligned -> b128 covers exactly)
    const float* s0 = conf + confbase;
    const int n128 = (min(TPB, Pq) * Cq) >> 2;
    for (int i = tid; i < n128; i += TPB)
      __builtin_amdgcn_global_load_async_to_lds_b128(
          (v4i*)(s0 + i * 4), (v4i*)(&s_conf[0][i * 4]), 0, 0);
  }
#endif

  for (int tle = 0; tle < ntiles; ++tle) {
    const int t0 = tle * TPB;
    const int cur = tle & 1;

#if USE_ASYNC
    __builtin_amdgcn_s_wait_asynccnt(0);   // my tile-`tle` DMAs landed
    __syncthreads();                       // everyone's landed; prev readers done
    if (tle + 1 < ntiles) {                // kick off next tile into other buffer
      const float* sn = conf + confbase + (size_t)(t0 + TPB) * Cq;
      const int n128 = (min(TPB, Pq - (t0 + TPB)) * Cq) >> 2;
      for (int i = tid; i < n128; i += TPB)
        __builtin_amdgcn_global_load_async_to_lds_b128(
            (v4i*)(sn + i * 4), (v4i*)(&s_conf[cur ^ 1][i * 4]), 0, 0);
    }
#else
    const int elems = min(TPB, Pq - t0) * Cq;
    const float* src = conf + confbase + (size_t)t0 * Cq;
    if (t0 + TPB < Pq)  // lowers to global_prefetch_b8 on gfx1250
      __builtin_prefetch(conf + confbase + (size_t)(t0 + TPB) * Cq + tid * 32, 0, 0);
    __syncthreads();
    for (int i = tid; i < elems; i += TPB) s_conf[cur][i] = src[i];
    __syncthreads();
#endif

    const int p = t0 + tid;
    const bool valid = (p < Pq);
    const int pc = valid ? p : (Pq - 1);

    // recompute this prior's best-truth match (cheap VALU; avoids 9MB spill)
    const float cx = dbox[pc*4+0], cy = dbox[pc*4+1];
    const float w  = dbox[pc*4+2], h  = dbox[pc*4+3];
    const float px0 = cx - 0.5f*w, py0 = cy - 0.5f*h;
    const float px1 = cx + 0.5f*w, py1 = cy + 0.5f*h;
    const float ab = w * h;
    float o = -1.f; int jb = 0;
#pragma unroll
    for (int j = 0; j < Oq; ++j) {
      const float tx0 = s_truth[j][0], ty0 = s_truth[j][1];
      const float tx1 = s_truth[j][2], ty1 = s_truth[j][3];
      const float iw = fmaxf(fminf(tx1,px1) - fmaxf(tx0,px0), 0.f);
      const float ih = fmaxf(fminf(ty1,py1) - fmaxf(ty0,py0), 0.f);
      const float inter = iw * ih;
      const float aa = (tx1-tx0) * (ty1-ty0);
      const float iou = inter / (aa + ab - inter);
      if (iou > o) { o = iou; jb = j; }
    }
    // forced-positive override (ascending t => last truth wins, as in JAX)
#pragma unroll
    for (int t = 0; t < Oq; ++t)
      if (p == s_bpidx[t]) { jb = t; o = 2.0f; }

    const int ct = (o < 0.5f) ? 0 : ((int)s_truth[jb][4] + 1);
    const bool pos = valid && (ct > 0);

    float* Row = &s_conf[cur][tid * Cq];
    float m = Row[0];
#pragma unroll
    for (int c = 1; c < Cq; ++c) m = fmaxf(m, Row[c]);
    const float xt = Row[ct];

#if USE_WMMA
    // overwrite consumed conf row with exp(x - m), in place (stride Cq)
#pragma unroll
    for (int c = 0; c < Cq; ++c) Row[c] = __expf(Row[c] - m);
    __syncthreads();

    // Sum_{c<21} exp via matrix pipe: D = A(16x4) * ones(4x16) accumulated
    // over 6 K-chunks (last chunk predicated to zero beyond c=20); every
    // column of D holds the row sum (fp32 exact).
    v8f acc0 = {0.f,0.f,0.f,0.f,0.f,0.f,0.f,0.f};
    v8f acc1 = {0.f,0.f,0.f,0.f,0.f,0.f,0.f,0.f};
    v2f bb; bb.x = 1.f; bb.y = 1.f;
    const float* Ew = &s_conf[cur][wave * 32 * Cq];
    const int M0 = lane & 15;
    const int kb = (lane >> 4) << 1;  // lanes 0-15: K 0,1 ; lanes 16-31: K 2,3
#pragma unroll
    for (int kc = 0; kc < 5; ++kc) {
      const int k = kc * 4 + kb;
      v2f a0; a0.x = Ew[M0 * Cq + k];        a0.y = Ew[M0 * Cq + k + 1];
      v2f a1; a1.x = Ew[(M0 + 16) * Cq + k]; a1.y = Ew[(M0 + 16) * Cq + k + 1];
      acc0 = __builtin_amdgcn_wmma_f32_16x16x4_f32(
          false, a0, false, bb, (short)0, acc0, false, false);
      acc1 = __builtin_amdgcn_wmma_f32_16x16x4_f32(
          false, a1, false, bb, (short)0, acc1, false, false);
    }
    {
      // K-chunk 5: k = 20..23; only k==20 (lanes 0-15, .x) is real data
      const bool lo = (lane < 16);
      v2f a0; a0.x = lo ? Ew[M0 * Cq + 20]        : 0.f; a0.y = 0.f;
      v2f a1; a1.x = lo ? Ew[(M0 + 16) * Cq + 20] : 0.f; a1.y = 0.f;
      acc0 = __builtin_amdgcn_wmma_f32_16x16x4_f32(
          false, a0, false, bb, (short)0, acc0, false, false);
      acc1 = __builtin_amdgcn_wmma_f32_16x16x4_f32(
          false, a1, false, bb, (short)0, acc1, false, false);
    }
    // D layout: VGPR r -> lanes 0-15 hold M=r, lanes 16-31 hold M=8+r
    if (lane == 0) {
#pragma unroll
      for (int r = 0; r < 8; ++r) {
        s_rsum[wave][r]      = acc0[r];
        s_rsum[wave][16 + r] = acc1[r];
      }
    }
    if (lane == 16) {
#pragma unroll
      for (int r = 0; r < 8; ++r) {
        s_rsum[wave][8 + r]  = acc0[r];
        s_rsum[wave][24 + r] = acc1[r];
      }
    }
    __syncthreads();
    const float se = s_rsum[wave][lane];
#else
    float se = 0.f;
#pragma unroll
    for (int c = 0; c < Cq; ++c) se += __expf(Row[c] - m);
#endif

    const float ce = m + __logf(se) - xt;
    if (valid) ceneg[(size_t)b * Pq + p] = pos ? 0.f : ce;

    if (pos) {
      cacc += ce;
      np += 1;
      const float mx0 = s_truth[jb][0], my0 = s_truth[jb][1];
      const float mx1 = s_truth[jb][2], my1 = s_truth[jb][3];
      const float gcx = ((mx0 + mx1) * 0.5f - cx) / (0.1f * w);
      const float gcy = ((my0 + my1) * 0.5f - cy) / (0.1f * h);
      const float gw  = __logf((mx1 - mx0) / w) / 0.2f;
      const float gh  = __logf((my1 - my0) / h) / 0.2f;
      const float* ld = loc + ((size_t)b * Pq + p) * 4;
      const float d0 = fabsf(ld[0] - gcx), d1 = fabsf(ld[1] - gcy);
      const float d2 = fabsf(ld[2] - gw),  d3 = fabsf(ld[3] - gh);
      lacc += (d0 < 1.f ? 0.5f*d0*d0 : d0 - 0.5f)
            + (d1 < 1.f ? 0.5f*d1*d1 : d1 - 0.5f)
            + (d2 < 1.f ? 0.5f*d2*d2 : d2 - 0.5f)
            + (d3 < 1.f ? 0.5f*d3*d3 : d3 - 0.5f);
    }
  }

  // -------- deterministic per-block reductions ------------------------------
  __syncthreads();
  s_red[tid] = lacc; __syncthreads();
  for (int s = TPB/2; s > 0; s >>= 1) { if (tid < s) s_red[tid] += s_red[tid+s]; __syncthreads(); }
  if (tid == 0) partL[b] = s_red[0];
  __syncthreads();
  s_red[tid] = cacc; __syncthreads();
  for (int s = TPB/2; s > 0; s >>= 1) { if (tid < s) s_red[tid] += s_red[tid+s]; __syncthreads(); }
  if (tid == 0) partC[b] = s_red[0];
  __syncthreads();
  s_red[tid] = (float)np; __syncthreads();
  for (int s = TPB/2; s > 0; s >>= 1) { if (tid < s) s_red[tid] += s_red[tid+s]; __syncthreads(); }
  if (tid == 0) numpos[b] = (int)(s_red[0] + 0.5f);
}

// ---------------------------------------------------------------------------
// Kernel 2: exact top-K sum of ce_neg per batch via 4-pass radix select.
// sum(topK) = sum(x > T) + (K - count(>T)) * T, with T the K-th largest.
// (ce_neg >= 0, so IEEE bit patterns are order-isomorphic to the values.)
// ---------------------------------------------------------------------------
__global__ __launch_bounds__(TPB) void mbx_negmine(
    const float* __restrict__ ceneg, const int* __restrict__ numpos,
    float* __restrict__ negC)
{
  const int b = blockIdx.x, tid = threadIdx.x;
  __shared__ unsigned s_hist[256];
  __shared__ int s_bin, s_kr;
  __shared__ float s_red[TPB];

  const int np = numpos[b];
  long long Kll = (long long)np * 3;
  if (Kll > Pq) Kll = Pq;
  if (Kll <= 0) { if (tid == 0) negC[b] = 0.f; return; }
  int Kr = (int)Kll;

  const float* cn = ceneg + (size_t)b * Pq;
  unsigned prefix = 0;

  for (int shift = 24; shift >= 0; shift -= 8) {
    s_hist[tid] = 0u;
    __syncthreads();
    const unsigned maskhi = (shift == 24) ? 0u : (0xFFFFFFFFu << (shift + 8));
    for (int p = tid; p < Pq; p += TPB) {
      const unsigned v = __float_as_uint(cn[p]);
      if ((v & maskhi) == prefix) atomicAdd(&s_hist[(v >> shift) & 255u], 1u);
    }
    __syncthreads();
    if (tid == 0) {
      unsigned cum = 0; int bin;
      for (bin = 255; bin >= 0; --bin) {
        const unsigned c = s_hist[bin];
        if (cum + c >= (unsigned)Kr) { s_bin = bin; s_kr = Kr - (int)cum; break; }
        cum += c;
      }
      if (bin < 0) { s_bin = 0; s_kr = 0; }
    }
    __syncthreads();
    prefix |= ((unsigned)s_bin) << shift;
    Kr = s_kr;
    __syncthreads();
  }

  const float T = __uint_as_float(prefix);
  float acc = 0.f;
  for (int p = tid; p < Pq; p += TPB) {
    const float v = cn[p];
    if (__float_as_uint(v) > prefix) acc += v;
  }
  s_red[tid] = acc; __syncthreads();
  for (int s = TPB/2; s > 0; s >>= 1) { if (tid < s) s_red[tid] += s_red[tid+s]; __syncthreads(); }
  if (tid == 0) negC[b] = s_red[0] + (float)Kr * T;
}

// ---------------------------------------------------------------------------
// Kernel 3: deterministic final reduction over batches + normalize by N.
// ---------------------------------------------------------------------------
__global__ __launch_bounds__(Bq) void mbx_fin(
    const float* __restrict__ partL, const float* __restrict__ partC,
    const float* __restrict__ negC, const int* __restrict__ numpos,
    float* __restrict__ out)
{
  __shared__ float rL[Bq], rC[Bq], rN[Bq];
  const int t = threadIdx.x;
  rL[t] = partL[t];
  rC[t] = partC[t] + negC[t];
  rN[t] = (float)numpos[t];
  __syncthreads();
  for (int s = Bq/2; s > 0; s >>= 1) {
    if (t < s) { rL[t] += rL[t+s]; rC[t] += rC[t+s]; rN[t] += rN[t+s]; }
    __syncthreads();
  }
  if (t == 0) { out[0] = rL[0] / rN[0]; out[1] = rC[0] / rN[0]; }
}

// ---------------------------------------------------------------------------
extern "C" void kernel_launch(void* const* d_in, const int* in_sizes, int n_in,
                              void* d_out, int out_size, void* d_ws, size_t ws_size,
                              hipStream_t stream) {
  const float* loc     = (const float*)d_in[0];
  const float* conf    = (const float*)d_in[1];
  const float* dbox    = (const float*)d_in[2];
  const float* targets = (const float*)d_in[3];
  float* out = (float*)d_out;

  char* ws = (char*)d_ws;
  float* partL  = (float*)(ws + 0);
  float* partC  = (float*)(ws + 512);
  float* negC   = (float*)(ws + 1024);
  int*   numpos = (int*)  (ws + 1536);
  float* ceneg  = (float*)(ws + 2048);   // B*P floats ~ 4.47 MB

  mbx_match_ce<<<Bq, TPB, 0, stream>>>(loc, conf, dbox, targets,
                                       partL, partC, numpos, ceneg);
  mbx_negmine<<<Bq, TPB, 0, stream>>>(ceneg, numpos, negC);
  mbx_fin<<<1, Bq, 0, stream>>>(partL, partC, negC, numpos, out);
}